// TuckerSpectralConv2d_31679678775903
// MI455X (gfx1250) — compile-verified
//
#include <hip/hip_runtime.h>

// ---------------------------------------------------------------------------
// TuckerSpectralConv2d for MI455X (gfx1250, wave32, WMMA).
//
// All FFTs are truncated-mode partial DFTs -> dense f32 GEMMs on the WMMA
// pipe (V_WMMA_F32_16X16X4_F32). Complex arithmetic is realized by Re/Im
// stacking so every heavy stage is a plain real GEMM. Panel staging into LDS
// uses the CDNA5 async global->LDS DMA path (ASYNCcnt).
// ---------------------------------------------------------------------------

typedef float v2f __attribute__((ext_vector_type(2)));
typedef float v8f __attribute__((ext_vector_type(8)));

#define TWO_PI_OVER_256 0.024543692606170259f

// ---------------- WMMA 16x16x4 f32 wrapper (compile-safe hedge) ------------
__device__ __forceinline__ v8f zero8() {
  v8f z;
#pragma unroll
  for (int i = 0; i < 8; ++i) z[i] = 0.0f;
  return z;
}

__device__ __forceinline__ v8f wmma4(v2f a, v2f b, v8f c) {
#if __has_builtin(__builtin_amdgcn_wmma_f32_16x16x4_f32)
  return __builtin_amdgcn_wmma_f32_16x16x4_f32(false, a, false, b, (short)0, c,
                                               false, false);
#else
  // Manual fallback with the same fragment layout (ISA 7.12.2).
  int lane = threadIdx.x & 31;
  int n = lane & 15;
  int mb = (lane >> 4) << 3;
#pragma unroll
  for (int v = 0; v < 8; ++v) {
    int mrow = mb + v;
    float sum = c[v];
#pragma unroll
    for (int k = 0; k < 4; ++k) {
      float av = __shfl((k & 1) ? a.y : a.x, mrow + ((k >> 1) << 4), 32);
      float bv = __shfl((k & 1) ? b.y : b.x, n + ((k >> 1) << 4), 32);
      sum += av * bv;
    }
    c[v] = sum;
  }
  return c;
#endif
}

// A fragment of a row-major MxK panel: lane (M%16 + 16*(K/2)), vgpr K%2.
__device__ __forceinline__ v2f load_a_frag(const float* A, int lda) {
  int lane = threadIdx.x & 31;
  int row = lane & 15;
  int kb = (lane >> 4) << 1;
  v2f a;
  a.x = A[row * lda + kb];
  a.y = A[row * lda + kb + 1];
  return a;
}
// B fragment of a row-major KxN panel: lane (N + 16*(K/2)), vgpr K%2.
__device__ __forceinline__ v2f load_b_frag(const float* B, int ldb) {
  int lane = threadIdx.x & 31;
  int col = lane & 15;
  int kb = (lane >> 4) << 1;
  v2f b;
  b.x = B[kb * ldb + col];
  b.y = B[(kb + 1) * ldb + col];
  return b;
}

// ---------------- CDNA5 async global->LDS DMA (ASYNCcnt) -------------------
// GLOBAL_LOAD_ASYNC_TO_LDS_B128 (GV mode): per-lane 16B copy, LDS byte
// address in a VGPR (low 32 bits of a generic LDS pointer == wave-relative
// LDS address per the flat-aperture truncation rule).
__device__ __forceinline__ void async_copy_b128(const float* lds_dst,
                                                const float* gsrc) {
  unsigned l = (unsigned)(size_t)lds_dst;
  asm volatile("global_load_async_to_lds_b128 %0, %1, off"
               :
               : "v"(l), "v"(gsrc)
               : "memory");
}
__device__ __forceinline__ void async_join() {
#if __has_builtin(__builtin_amdgcn_s_wait_asynccnt)
  __builtin_amdgcn_s_wait_asynccnt(0);
#else
  asm volatile("s_wait_asynccnt 0x0" ::: "memory");
#endif
}
// Cooperative contiguous panel copy: nfloats must be a multiple of 4*blockDim.
__device__ __forceinline__ void stage_panel_async(float* lds, const float* g,
                                                  int nfloats) {
  for (int i = threadIdx.x * 4; i < nfloats; i += blockDim.x * 4)
    async_copy_b128(lds + i, g + i);
  async_join();
  __syncthreads();
}

// ---------------------------------------------------------------------------
#define NB 8
#define NH 256
#define NW 256

// Workspace layout (in floats)
#define OFF_BF 0u            // Bf_cat  256x64   [B_f.re | B_f.im]
#define OFF_MA 16384u        // MA      64x512   [[Ar|-Ai],[Ai|Ar]]
#define OFF_AI 49152u        // Ai_cat  256x64   [Ar | Ai]
#define OFF_BST 65536u       // Bstack  64x256   [Bi_r ; -Bi_i]
#define OFF_G1 81920u        // G1 planar: re 131072 + im 131072
#define OFF_GC 344064u       // Gcat    4096x128 [G2r | G2i]
#define OFF_SB 868352u       // Sbig    128x2048 [[Sr|Si],[-Si|Sr]]
#define OFF_WR 1130496u      // Wr      1024x4096 (mode-major)
#define OFF_WI 5324800u      // Wi      1024x4096
#define OFF_XL 9519104u      // Xlow    512 x [2][32][32]
#define OFF_YL 10567680u     // Ylow    512 x [2][32][32]
#define OFF_T1 11616256u     // T1 / Pcat (reused): 512 x 16384
// total = 20,004,864 floats = 80.02 MB

// ---------------------------------------------------------------------------
// Kernel 1: DFT basis matrices. rfft2/irfft2 'ortho' => 1/16 per axis.
// ---------------------------------------------------------------------------
__global__ void kbasis(float* __restrict__ Bf, float* __restrict__ MA,
                       float* __restrict__ Ai, float* __restrict__ Bst) {
  int t = blockIdx.x * blockDim.x + threadIdx.x;
  if (t < 16384) {  // Bf_cat[n][j]: e^{-2pi i n ky/256}/16
    int n = t >> 6, j = t & 63;
    int ky = j & 31;
    float th = TWO_PI_OVER_256 * (float)((n * ky) & 255);
    Bf[t] = (j < 32 ? cosf(th) : -sinf(th)) * 0.0625f;
  } else if (t < 49152) {  // MA[r][c] from A_f = e^{-2pi i kx m/256}/16
    int i = t - 16384;
    int r = i >> 9, c = i & 511;
    int kx = r & 31, m = c & 255;
    float th = TWO_PI_OVER_256 * (float)((kx * m) & 255);
    float re = cosf(th) * 0.0625f;
    float im = -sinf(th) * 0.0625f;
    MA[i] = (r < 32) ? ((c < 256) ? re : -im) : ((c < 256) ? im : re);
  } else if (t < 65536) {  // Ai_cat[m][j]: e^{+2pi i kx m/256}/16
    int i = t - 49152;
    int m = i >> 6, j = i & 63;
    int kx = j & 31;
    float th = TWO_PI_OVER_256 * (float)((kx * m) & 255);
    Ai[i] = (j < 32 ? cosf(th) : sinf(th)) * 0.0625f;
  } else if (t < 81920) {  // Bstack[r][n]: alpha*e^{+2pi i ky n/256}/16
    int i = t - 65536;
    int r = i >> 8, n = i & 255;
    int ky = r & 31;
    float alpha = (ky == 0) ? 1.0f : 2.0f;  // Hermitian doubling for irfft
    float th = TWO_PI_OVER_256 * (float)((ky * n) & 255);
    Bst[i] = (r < 32 ? cosf(th) : -sinf(th)) * 0.0625f * alpha;
  }
}

// ---------------------------------------------------------------------------
// Kernel 2a: G1[i,k,l] = sum_j U[i,j]*G[j,k,l]
// ---------------------------------------------------------------------------
__global__ void kG1(const float2* __restrict__ U, const float2* __restrict__ G,
                    float* __restrict__ G1r, float* __restrict__ G1i) {
  int t = blockIdx.x * blockDim.x + threadIdx.x;  // 131072
  int i = t >> 11;
  int rem = t & 2047;
  int k = rem >> 6, l = rem & 63;
  float ar = 0.f, ai = 0.f;
#pragma unroll 4
  for (int j = 0; j < 32; ++j) {
    float2 u = U[i * 32 + j];
    float2 g = G[(j * 32 + k) * 64 + l];
    ar += u.x * g.x - u.y * g.y;
    ai += u.x * g.y + u.y * g.x;
  }
  G1r[t] = ar;
  G1i[t] = ai;
}

// ---------------------------------------------------------------------------
// Kernel 2b: G2[i,o,l] = sum_k V[o,k]*G1[i,k,l] -> Gcat (4096 x 128) [re|im]
// ---------------------------------------------------------------------------
__global__ void kG2(const float2* __restrict__ V, const float* __restrict__ G1r,
                    const float* __restrict__ G1i, float* __restrict__ Gc) {
  int t = blockIdx.x * blockDim.x + threadIdx.x;  // 262144
  int i = t >> 12;
  int rem = t & 4095;
  int o = rem >> 6, l = rem & 63;
  float ar = 0.f, ai = 0.f;
#pragma unroll 4
  for (int k = 0; k < 32; ++k) {
    float2 v = V[o * 32 + k];
    float gr = G1r[i * 2048 + k * 64 + l];
    float gi = G1i[i * 2048 + k * 64 + l];
    ar += v.x * gr - v.y * gi;
    ai += v.x * gi + v.y * gr;
  }
  int row = i * 64 + o;
  Gc[row * 128 + l] = ar;
  Gc[row * 128 + 64 + l] = ai;
}

// ---------------------------------------------------------------------------
// Kernel 2c: Sbig (128 x 2048) = [[Sr | Si],[-Si | Sr]] from S^T (l, m)
// ---------------------------------------------------------------------------
__global__ void kSbig(const float2* __restrict__ S, float* __restrict__ Sb) {
  int t = blockIdx.x * blockDim.x + threadIdx.x;  // 262144
  int r = t >> 11;
  int c = t & 2047;
  int l = r & 63;
  int m = c & 1023;
  float2 s = S[(size_t)m * 64 + l];
  float v;
  if (c < 1024)
    v = (r < 64) ? s.x : -s.y;
  else
    v = (r < 64) ? s.y : s.x;
  Sb[t] = v;
}

// ---------------------------------------------------------------------------
// Forward stage 1: per (b,c) image, T1cat(256x64) = x(256x256) @ Bf_cat.
// B panel async-staged into 64KB LDS; A stream prefetched one k-block ahead.
// ---------------------------------------------------------------------------
__global__ void kfwd1(const float* __restrict__ x, const float* __restrict__ Bf,
                      float* __restrict__ T1) {
  __shared__ __align__(16) float sB[256 * 64];
  int bc = blockIdx.x;
  const float* xi = x + (size_t)bc * (NH * NW);
  float* T1o = T1 + (size_t)bc * 16384;
  stage_panel_async(sB, Bf, 256 * 64);

  int wave = threadIdx.x >> 5;
  int lane = threadIdx.x & 31;
  int n0 = lane & 15, mb = (lane >> 4) << 3;

  for (int mi = 0; mi < 2; ++mi) {
    int mt = wave * 2 + mi;  // 16 M tiles over 8 waves
    const float* Ap = xi + (mt * 16) * 256;
    v8f acc[4];
#pragma unroll
    for (int nt = 0; nt < 4; ++nt) acc[nt] = zero8();
    for (int k = 0; k < 256; k += 4) {
      if ((k & 63) == 0 && k < 192)  // prefetch next 64-col block of A rows
        __builtin_prefetch(Ap + (lane & 15) * 256 + k + 64, 0, 1);
      v2f a = load_a_frag(Ap + k, 256);
#pragma unroll
      for (int nt = 0; nt < 4; ++nt) {
        v2f b = load_b_frag(sB + k * 64 + nt * 16, 64);
        acc[nt] = wmma4(a, b, acc[nt]);
      }
    }
#pragma unroll
    for (int nt = 0; nt < 4; ++nt) {
#pragma unroll
      for (int v = 0; v < 8; ++v) {
        int row = mt * 16 + mb + v;
        int col = nt * 16 + n0;
        T1o[(col >> 5) * 8192 + row * 32 + (col & 31)] = acc[nt][v];
      }
    }
  }
}

// ---------------------------------------------------------------------------
// Forward stage 2: per (b,c): Out(64x32) = MA(64x512) @ Tstack(512x32).
// ---------------------------------------------------------------------------
__global__ void kfwd2(const float* __restrict__ MA, const float* __restrict__ T1,
                      float* __restrict__ Xlow) {
  int bc = blockIdx.x;
  const float* T = T1 + (size_t)bc * 16384;  // viewed as 512x32
  float* Xo = Xlow + (size_t)bc * 2048;
  int wave = threadIdx.x >> 5;
  int mt = wave >> 1, nt = wave & 1;
  v8f acc = zero8();
  for (int k = 0; k < 512; k += 4) {
    v2f a = load_a_frag(MA + mt * 16 * 512 + k, 512);
    v2f b = load_b_frag(T + k * 32 + nt * 16, 32);
    acc = wmma4(a, b, acc);
  }
  int lane = threadIdx.x & 31;
  int n0 = lane & 15, mb = (lane >> 4) << 3;
#pragma unroll
  for (int v = 0; v < 8; ++v) {
    int row = mt * 16 + mb + v;  // 0..63 (re/im stacked)
    int col = nt * 16 + n0;      // 0..31
    Xo[(row >> 5) * 1024 + (row & 31) * 32 + col] = acc[v];
  }
}

// ---------------------------------------------------------------------------
// W_hat GEMM: Out(4096 x 2048) = Gcat(4096x128) @ Sbig(128x2048).
// All 8 waves of a block share one async-staged A panel (16x128, 8KB LDS).
// Result tiles bounce through LDS so the mode-major stores are 64B-contiguous
// (consecutive lanes -> consecutive rows of Wr/Wi planes).
// ---------------------------------------------------------------------------
__global__ void kwhat(const float* __restrict__ Gc, const float* __restrict__ Sb,
                      float* __restrict__ Wr, float* __restrict__ Wi) {
  __shared__ __align__(16) float sA[16 * 128];
  __shared__ float sT[8][16 * 16];
  int wave = threadIdx.x >> 5;
  int lane = threadIdx.x & 31;
  int tm = blockIdx.x >> 4;                  // 0..255
  int tn = (blockIdx.x & 15) * 8 + wave;     // 0..127

  stage_panel_async(sA, Gc + (size_t)tm * 16 * 128, 16 * 128);

  v8f acc = zero8();
  for (int k = 0; k < 128; k += 4) {
    v2f a = load_a_frag(sA + k, 128);
    v2f b = load_b_frag(Sb + k * 2048 + tn * 16, 2048);
    acc = wmma4(a, b, acc);
  }
  // park the 16x16 tile in this wave's LDS slab
  int n0 = lane & 15, mb = (lane >> 4) << 3;
  float* t = sT[wave];
#pragma unroll
  for (int v = 0; v < 8; ++v) t[(mb + v) * 16 + n0] = acc[v];
  __syncthreads();
  // transposed, coalesced store; a 16-wide N tile never straddles Re|Im.
  float* dst = (tn < 64) ? Wr : Wi;
  int colbase = (tn & 63) * 16;
  int rowbase = tm * 16;
  int row = lane & 15;
  int ch = lane >> 4;
#pragma unroll
  for (int cc = 0; cc < 8; ++cc) {
    int col = cc * 2 + ch;
    dst[(size_t)(colbase + col) * 4096 + rowbase + row] = t[row * 16 + col];
  }
}

// ---------------------------------------------------------------------------
// Mode mixing: per mode m, Y(8x128) = Xcat(8x128) @ Wbig(128x128),
// Wbig = [[Wr, Wi],[-Wi, Wr]] in 64KB LDS.  M padded to 16 for WMMA.
// ---------------------------------------------------------------------------
__device__ __forceinline__ float xmix_load(const float* __restrict__ Xl, int b,
                                           int kk, int m) {
  if (b >= 8) return 0.0f;  // M padding rows
  int ch = kk & 63, plane = kk >> 6;
  return Xl[(size_t)(b * 64 + ch) * 2048 + plane * 1024 + m];
}

__global__ void kmix(const float* __restrict__ Xlow, const float* __restrict__ Wr,
                     const float* __restrict__ Wi, float* __restrict__ Ylow) {
  __shared__ float sW[128 * 128];
  int m = blockIdx.x;  // mode = x*32+y == plane offset
  const float* wr = Wr + (size_t)m * 4096;
  const float* wi = Wi + (size_t)m * 4096;
  for (int i = threadIdx.x; i < 16384; i += blockDim.x) {
    int r = i >> 7, c = i & 127;
    int i0 = r & 63, o0 = c & 63;
    float v;
    if (r < 64)
      v = (c < 64) ? wr[i0 * 64 + o0] : wi[i0 * 64 + o0];
    else
      v = (c < 64) ? -wi[i0 * 64 + o0] : wr[i0 * 64 + o0];
    sW[i] = v;
  }
  __syncthreads();

  int nt = threadIdx.x >> 5;  // 8 N tiles
  int lane = threadIdx.x & 31;
  int arow = lane & 15;
  int akb = (lane >> 4) << 1;
  v8f acc = zero8();
  for (int k = 0; k < 128; k += 4) {
    v2f a;
    a.x = xmix_load(Xlow, arow, k + akb, m);
    a.y = xmix_load(Xlow, arow, k + akb + 1, m);
    v2f b = load_b_frag(sW + k * 128 + nt * 16, 128);
    acc = wmma4(a, b, acc);
  }
  int n0 = lane & 15, mb = (lane >> 4) << 3;
#pragma unroll
  for (int v = 0; v < 8; ++v) {
    int row = mb + v;  // batch b
    int col = nt * 16 + n0;
    if (row < 8) {
      int o = col & 63, plane = col >> 6;
      Ylow[(size_t)(row * 64 + o) * 2048 + plane * 1024 + m] = acc[v];
    }
  }
}

// ---------------------------------------------------------------------------
// Inverse stage 1: per (b,o): Pcat(256x64) = Ai_cat(256x64) @ Ybig(64x64),
// Ybig = [[Yr, Yi],[-Yi, Yr]] built in LDS.
// ---------------------------------------------------------------------------
__global__ void kinv1(const float* __restrict__ Ai, const float* __restrict__ Ylow,
                      float* __restrict__ Pcat) {
  __shared__ float sY[64 * 64];
  int bo = blockIdx.x;
  const float* Yr = Ylow + (size_t)bo * 2048;
  const float* Yi = Yr + 1024;
  for (int i = threadIdx.x; i < 4096; i += blockDim.x) {
    int r = i >> 6, c = i & 63;
    int r0 = r & 31, c0 = c & 31;
    float v;
    if (r < 32)
      v = (c < 32) ? Yr[r0 * 32 + c0] : Yi[r0 * 32 + c0];
    else
      v = (c < 32) ? -Yi[r0 * 32 + c0] : Yr[r0 * 32 + c0];
    sY[i] = v;
  }
  __syncthreads();

  float* P = Pcat + (size_t)bo * 16384;
  int wave = threadIdx.x >> 5;
  int lane = threadIdx.x & 31;
  int n0 = lane & 15, mb = (lane >> 4) << 3;
  for (int mi = 0; mi < 2; ++mi) {
    int mt = wave * 2 + mi;
    v8f acc[4];
#pragma unroll
    for (int nt = 0; nt < 4; ++nt) acc[nt] = zero8();
    for (int k = 0; k < 64; k += 4) {
      v2f a = load_a_frag(Ai + mt * 16 * 64 + k, 64);
#pragma unroll
      for (int nt = 0; nt < 4; ++nt) {
        v2f b = load_b_frag(sY + k * 64 + nt * 16, 64);
        acc[nt] = wmma4(a, b, acc[nt]);
      }
    }
#pragma unroll
    for (int nt = 0; nt < 4; ++nt) {
#pragma unroll
      for (int v = 0; v < 8; ++v) {
        int row = mt * 16 + mb + v;
        int col = nt * 16 + n0;
        P[row * 64 + col] = acc[nt][v];
      }
    }
  }
}

// ---------------------------------------------------------------------------
// Inverse stage 2: per (b,o): y(256x256) = Pcat(256x64) @ Bstack(64x256).
// [Pr|Pi] @ [Bi_r; -Bi_i] == Re(P @ B_i) -> real output directly.
// B panel async-staged into 64KB LDS; 16 accumulator tiles/wave for A reuse.
// ---------------------------------------------------------------------------
__global__ void kinv2(const float* __restrict__ Pcat, const float* __restrict__ Bst,
                      float* __restrict__ y) {
  __shared__ __align__(16) float sB[64 * 256];
  int bo = blockIdx.x;
  const float* P = Pcat + (size_t)bo * 16384;
  float* yo = y + (size_t)bo * (NH * NW);
  stage_panel_async(sB, Bst, 64 * 256);

  int wave = threadIdx.x >> 5;
  int lane = threadIdx.x & 31;
  int n0 = lane & 15, mb = (lane >> 4) << 3;
  for (int mi = 0; mi < 2; ++mi) {
    int mt = wave * 2 + mi;
    v8f acc[16];
#pragma unroll
    for (int nt = 0; nt < 16; ++nt) acc[nt] = zero8();
    for (int k = 0; k < 64; k += 4) {
      v2f a = load_a_frag(P + mt * 16 * 64 + k, 64);
#pragma unroll
      for (int nt = 0; nt < 16; ++nt) {
        v2f b = load_b_frag(sB + k * 256 + nt * 16, 256);
        acc[nt] = wmma4(a, b, acc[nt]);
      }
    }
#pragma unroll
    for (int nt = 0; nt < 16; ++nt) {
#pragma unroll
      for (int v = 0; v < 8; ++v) {
        int row = mt * 16 + mb + v;
        int col = nt * 16 + n0;
        yo[row * 256 + col] = acc[nt][v];
      }
    }
  }
}

// ---------------------------------------------------------------------------
// Host launch
// ---------------------------------------------------------------------------
extern "C" void kernel_launch(void* const* d_in, const int* in_sizes, int n_in,
                              void* d_out, int out_size, void* d_ws,
                              size_t ws_size, hipStream_t stream) {
  (void)in_sizes; (void)n_in; (void)out_size; (void)ws_size;
  const float*  x = (const float*)d_in[0];
  const float2* U = (const float2*)d_in[1];
  const float2* V = (const float2*)d_in[2];
  const float2* S = (const float2*)d_in[3];
  const float2* G = (const float2*)d_in[4];
  float* y = (float*)d_out;
  float* w = (float*)d_ws;

  float* Bf  = w + OFF_BF;
  float* MA  = w + OFF_MA;
  float* Ai  = w + OFF_AI;
  float* Bst = w + OFF_BST;
  float* G1r = w + OFF_G1;
  float* G1i = w + OFF_G1 + 131072u;
  float* Gc  = w + OFF_GC;
  float* Sb  = w + OFF_SB;
  float* Wr  = w + OFF_WR;
  float* Wi  = w + OFF_WI;
  float* Xl  = w + OFF_XL;
  float* Yl  = w + OFF_YL;
  float* T1  = w + OFF_T1;   // also reused as Pcat (disjoint phases)

  const dim3 blk(256);
  kbasis<<<dim3(320),  blk, 0, stream>>>(Bf, MA, Ai, Bst);
  kG1   <<<dim3(512),  blk, 0, stream>>>(U, G, G1r, G1i);
  kG2   <<<dim3(1024), blk, 0, stream>>>(V, G1r, G1i, Gc);
  kSbig <<<dim3(1024), blk, 0, stream>>>(S, Sb);
  kfwd1 <<<dim3(512),  blk, 0, stream>>>(x, Bf, T1);
  kfwd2 <<<dim3(512),  blk, 0, stream>>>(MA, T1, Xl);
  kwhat <<<dim3(4096), blk, 0, stream>>>(Gc, Sb, Wr, Wi);
  kmix  <<<dim3(1024), blk, 0, stream>>>(Xl, Wr, Wi, Yl);
  kinv1 <<<dim3(512),  blk, 0, stream>>>(Ai, Yl, T1 /*Pcat*/);
  kinv2 <<<dim3(512),  blk, 0, stream>>>(T1 /*Pcat*/, Bst, y);
}